// MetaLearner_86208583565567
// MI455X (gfx1250) — compile-verified
//
#include <hip/hip_runtime.h>

typedef __attribute__((ext_vector_type(2))) float v2f;
typedef __attribute__((ext_vector_type(8))) float v8f;

#define T_STEPS 10
#define N_BATCH 256
#define DIN 1024
#define DOUT 1000
#define UNITS 20
#define NW (DIN * DOUT)        /* 1,024,000 */
#define NP (NW + DOUT)         /* 1,025,000 */
#define NPP 1025008            /* padded to multiple of 16 */
#define NT_COLS 63             /* ceil(1000/16) */
#define P_INV 0.1f
#define EXP_NEG_P 4.5399929762484854e-05f
#define EXP_P 22026.465794806718f

__device__ __forceinline__ float hsig(float x) { return fminf(fmaxf(0.2f * x + 0.5f, 0.f), 1.f); }
__device__ __forceinline__ float sigm(float x) { return 1.f / (1.f + expf(-x)); }

__device__ __forceinline__ void preprocess(float x, float &a, float &b) {
  float ax = fabsf(x);
  float m1 = (ax > EXP_NEG_P) ? 1.f : 0.f;
  float m2 = 1.f - m1;
  a = m1 * logf(ax + m2) * P_INV - m2;
  float sgn = (x > 0.f) ? 1.f : ((x < 0.f) ? -1.f : 0.f);
  b = m1 * sgn + m2 * EXP_P * x;
}

// logits(256,1000) = x(256,1024) @ W(1024,1000) + b ; W,b packed in p
__global__ void k_gemm_logits(const float* __restrict__ x,
                              const float* __restrict__ p,
                              float* __restrict__ out) {
  int wave = blockIdx.x * 4 + (threadIdx.x >> 5);
  int lane = threadIdx.x & 31;
  int mt = wave / NT_COLS, nt = wave % NT_COLS;
  int mn = lane & 15, grp = lane >> 4;
  int col = nt * 16 + mn;
  int colc = (col < DOUT) ? col : (DOUT - 1);
  const float* __restrict__ arow = x + (size_t)(mt * 16 + mn) * DIN;
  v8f acc = {};
  for (int k = 0; k < DIN; k += 4) {
    int kk = k + 2 * grp;
    v2f a, b;
    a.x = arow[kk];
    a.y = arow[kk + 1];
    b.x = p[(size_t)kk * DOUT + colc];
    b.y = p[(size_t)(kk + 1) * DOUT + colc];
    acc = __builtin_amdgcn_wmma_f32_16x16x4_f32(false, a, false, b, (short)0, acc, false, false);
  }
  if (col < DOUT) {
    float bv = p[NW + col];
#pragma unroll
    for (int r = 0; r < 8; ++r) {
      int row = mt * 16 + r + 8 * grp;
      out[(size_t)row * DOUT + col] = acc[r] + bv;
    }
  }
}

// per-row softmax; dlogits = softmax - y in place; loss += sum(y*(logZ - logit))
__global__ void k_softmax(float* __restrict__ dl, const float* __restrict__ y,
                          float* __restrict__ loss) {
  __shared__ float red[256];
  int r = blockIdx.x, tid = threadIdx.x;
  float* __restrict__ lg = dl + (size_t)r * DOUT;
  const float* __restrict__ yr = y + (size_t)r * DOUT;
  float v = -3.4e38f;
  for (int j = tid; j < DOUT; j += 256) v = fmaxf(v, lg[j]);
  red[tid] = v; __syncthreads();
  for (int s = 128; s > 0; s >>= 1) { if (tid < s) red[tid] = fmaxf(red[tid], red[tid + s]); __syncthreads(); }
  float mx = red[0]; __syncthreads();
  float se = 0.f;
  for (int j = tid; j < DOUT; j += 256) se += expf(lg[j] - mx);
  red[tid] = se; __syncthreads();
  for (int s = 128; s > 0; s >>= 1) { if (tid < s) red[tid] += red[tid + s]; __syncthreads(); }
  float Z = red[0]; __syncthreads();
  float logZ = mx + logf(Z);
  float ll = 0.f;
  for (int j = tid; j < DOUT; j += 256) {
    float l = lg[j];
    float yv = yr[j];
    ll += yv * (logZ - l);
    lg[j] = expf(l - mx) / Z - yv;
  }
  red[tid] = ll; __syncthreads();
  for (int s = 128; s > 0; s >>= 1) { if (tid < s) red[tid] += red[tid + s]; __syncthreads(); }
  if (tid == 0) atomicAdd(loss, red[0]);
}

// gb[o] = sum_r dlogits[r][o]  -> g[NW + o]
__global__ void k_gb(const float* __restrict__ dl, float* __restrict__ g) {
  int o = blockIdx.x * 256 + threadIdx.x;
  if (o >= DOUT) return;
  float s = 0.f;
  for (int r = 0; r < N_BATCH; ++r) s += dl[(size_t)r * DOUT + o];
  g[NW + o] = s;
}

// gW(1024,1000) = x^T(1024,256) @ dlogits(256,1000) -> g[0..NW)
__global__ void k_gemm_gw(const float* __restrict__ x,
                          const float* __restrict__ dl,
                          float* __restrict__ g) {
  int wave = blockIdx.x * 4 + (threadIdx.x >> 5);
  int lane = threadIdx.x & 31;
  int dt = wave / NT_COLS, nt = wave % NT_COLS;
  int mn = lane & 15, grp = lane >> 4;
  int col = nt * 16 + mn;
  int colc = (col < DOUT) ? col : (DOUT - 1);
  int d = dt * 16 + mn;
  v8f acc = {};
  for (int k = 0; k < N_BATCH; k += 4) {
    int kk = k + 2 * grp;
    v2f a, b;
    a.x = x[(size_t)kk * DIN + d];
    a.y = x[(size_t)(kk + 1) * DIN + d];
    b.x = dl[(size_t)kk * DOUT + colc];
    b.y = dl[(size_t)(kk + 1) * DOUT + colc];
    acc = __builtin_amdgcn_wmma_f32_16x16x4_f32(false, a, false, b, (short)0, acc, false, false);
  }
  if (col < DOUT) {
#pragma unroll
    for (int r = 0; r < 8; ++r) {
      int row = dt * 16 + r + 8 * grp;
      g[(size_t)row * DOUT + col] = acc[r];
    }
  }
}

// Coordinatewise LSTM + param update; one wave per 16 params.
// h,c are unit-major: h[unit*NPP + param] -> coalesced WMMA A-frag loads.
__global__ void k_lstm(float* __restrict__ pbuf, const float* __restrict__ gbuf,
                       float* __restrict__ hbuf, float* __restrict__ cbuf,
                       float* __restrict__ fbuf, float* __restrict__ ibuf,
                       const float* __restrict__ lossp,
                       const float* __restrict__ kern,   // (5,80)
                       const float* __restrict__ rk,     // (20,80)
                       const float* __restrict__ bias,   // (80)
                       const float* __restrict__ Wf, const float* __restrict__ bf,
                       const float* __restrict__ Wi, const float* __restrict__ bi) {
  __shared__ float z_lds[8][16][80];
  __shared__ float h_lds[8][16][UNITS];
  __shared__ float pa[8][16], q1a[8][16], q2a[8][16];

  int wv = threadIdx.x >> 5;
  int lane = threadIdx.x & 31;
  long wg = (long)blockIdx.x * 8 + wv;
  if (wg >= (NPP / 16)) return;           // divergent wave exit is safe: no block barriers
  long pb = wg * 16;
  int mn = lane & 15, grp = lane >> 4;

  float lossv = lossp[0];
  float pl1, pl2;
  preprocess(lossv, pl1, pl2);

  float p_loc = 0.f, g_loc = 0.f;
  if (lane < 16) {
    long idx = pb + lane;
    p_loc = pbuf[idx];
    g_loc = gbuf[idx];
    float a1, a2;
    preprocess(g_loc, a1, a2);
    pa[wv][lane] = p_loc; q1a[wv][lane] = a1; q2a[wv][lane] = a2;
  }
  __builtin_amdgcn_wave_barrier();

  float pv[8], g1v[8], g2v[8];
#pragma unroll
  for (int r = 0; r < 8; ++r) {
    pv[r]  = pa[wv][8 * grp + r];
    g1v[r] = q1a[wv][8 * grp + r];
    g2v[r] = q2a[wv][8 * grp + r];
  }

#pragma unroll
  for (int nt = 0; nt < 5; ++nt) {
    int col = nt * 16 + mn;
    v8f acc = {};
#pragma unroll
    for (int ks = 0; ks < UNITS; ks += 4) {
      int kk = ks + 2 * grp;
      v2f a, b;
      a.x = hbuf[(size_t)kk * NPP + pb + mn];
      a.y = hbuf[(size_t)(kk + 1) * NPP + pb + mn];
      b.x = rk[kk * 80 + col];
      b.y = rk[(kk + 1) * 80 + col];
      acc = __builtin_amdgcn_wmma_f32_16x16x4_f32(false, a, false, b, (short)0, acc, false, false);
    }
    float base = bias[col] + pl1 * kern[3 * 80 + col] + pl2 * kern[4 * 80 + col];
    float k0c = kern[0 * 80 + col], k1c = kern[1 * 80 + col], k2c = kern[2 * 80 + col];
#pragma unroll
    for (int r = 0; r < 8; ++r) {
      int pr = r + 8 * grp;
      z_lds[wv][pr][col] = acc[r] + base + pv[r] * k0c + g1v[r] * k1c + g2v[r] * k2c;
    }
  }
  __builtin_amdgcn_wave_barrier();

#pragma unroll
  for (int it = 0; it < 10; ++it) {
    int unit = it * 2 + grp;   // 0..19
    int prm = mn;
    float zi = z_lds[wv][prm][unit];
    float zf = z_lds[wv][prm][UNITS + unit];
    float zc = z_lds[wv][prm][2 * UNITS + unit];
    float zo = z_lds[wv][prm][3 * UNITS + unit];
    size_t gidx = (size_t)unit * NPP + pb + prm;
    float co = cbuf[gidx];
    float cn = hsig(zf) * co + hsig(zi) * tanhf(zc);
    float hn = hsig(zo) * tanhf(cn);
    cbuf[gidx] = cn;
    hbuf[gidx] = hn;
    h_lds[wv][prm][unit] = hn;
  }
  __builtin_amdgcn_wave_barrier();

  if (lane < 16) {
    long idx = pb + lane;
    float fo = fbuf[idx], io = ibuf[idx];
    float df = bf[0], di = bi[0];
#pragma unroll
    for (int u = 0; u < UNITS; ++u) {
      float hv = h_lds[wv][lane][u];
      df += hv * Wf[u];
      di += hv * Wi[u];
    }
    df += fo * Wf[UNITS];
    di += io * Wi[UNITS];
    float fn = sigm(df), inn = sigm(di);
    fbuf[idx] = fn;
    ibuf[idx] = inn;
    pbuf[idx] = fn * p_loc - inn * g_loc;
  }
}

extern "C" void kernel_launch(void* const* d_in, const int* in_sizes, int n_in,
                              void* d_out, int out_size, void* d_ws, size_t ws_size,
                              hipStream_t stream) {
  (void)in_sizes; (void)n_in; (void)out_size; (void)ws_size;
  const float* feats  = (const float*)d_in[0];
  const float* labels = (const float*)d_in[1];
  const float* params = (const float*)d_in[2];
  const float* kern   = (const float*)d_in[3];
  const float* rk     = (const float*)d_in[4];
  const float* bias   = (const float*)d_in[5];
  const float* Wf     = (const float*)d_in[6];
  const float* bf     = (const float*)d_in[7];
  const float* Wi     = (const float*)d_in[8];
  const float* bi     = (const float*)d_in[9];

  float* ws = (float*)d_ws;
  float* h  = ws;
  float* c  = h + (size_t)UNITS * NPP;
  float* p  = c + (size_t)UNITS * NPP;
  float* f  = p + NPP;
  float* iv = f + NPP;
  float* g  = iv + NPP;
  float* dl = g + NPP;
  float* loss = dl + (size_t)N_BATCH * DOUT;

  // deterministic re-init every call
  hipMemsetAsync(h, 0, (size_t)UNITS * NPP * sizeof(float), stream);
  hipMemsetAsync(c, 0, (size_t)UNITS * NPP * sizeof(float), stream);
  hipMemsetAsync(f, 0, (size_t)NPP * sizeof(float), stream);
  hipMemsetAsync(iv, 0, (size_t)NPP * sizeof(float), stream);
  hipMemsetAsync(g + NP, 0, (size_t)(NPP - NP) * sizeof(float), stream);
  hipMemsetAsync(p + NP, 0, (size_t)(NPP - NP) * sizeof(float), stream);
  hipMemcpyAsync(p, params, (size_t)NP * sizeof(float), hipMemcpyDeviceToDevice, stream);

  for (int t = 0; t < T_STEPS; ++t) {
    const float* xt = feats + (size_t)t * N_BATCH * DIN;
    const float* yt = labels + (size_t)t * N_BATCH * DOUT;
    hipMemsetAsync(loss, 0, sizeof(float), stream);
    k_gemm_logits<<<252, 128, 0, stream>>>(xt, p, dl);        // 16*63 wave tiles
    k_softmax<<<N_BATCH, 256, 0, stream>>>(dl, yt, loss);
    k_gb<<<4, 256, 0, stream>>>(dl, g);
    k_gemm_gw<<<1008, 128, 0, stream>>>(xt, dl, g);           // 64*63 wave tiles
    k_lstm<<<8008, 256, 0, stream>>>(p, g, h, c, f, iv, loss,
                                     kern, rk, bias, Wf, bf, Wi, bi);
  }
  hipMemcpyAsync(d_out, p, (size_t)NP * sizeof(float), hipMemcpyDeviceToDevice, stream);
}